// GCwithself_14250701488882
// MI455X (gfx1250) — compile-verified
//
#include <hip/hip_runtime.h>
#include <hip/hip_bf16.h>

// D = 128 fixed feature dim.
typedef __attribute__((ext_vector_type(2))) float v2f;
typedef __attribute__((ext_vector_type(8))) float v8f;

#define DFEAT 128

// ---------------------------------------------------------------------------
// Kernel 1: dual GEMM via V_WMMA_F32_16X16X4_F32.
//   y[n,d]   = sum_k x[n,k]*W[d,k]      + b[d]      -> d_ws
//   out[n,d] = sum_k x[n,k]*W_self[d,k] + b_self[d] -> d_out (pre-relu base)
// Block = 256 threads = 8 waves. blockIdx.x = 16-row tile. Wave w owns the
// 16-column tile [16w, 16w+16) of D=128.
//
// WMMA f32 16x16x4 lane layouts (ISA 7.12.2, 32-bit A 16x4 / C,D 16x16):
//   A : lane<16 -> M=lane,    VGPR0=K0, VGPR1=K1 ; lane>=16 -> M=lane-16, K2/K3
//   B : lane<16 -> N=lane,    VGPR0=K0, VGPR1=K1 ; lane>=16 -> N=lane-16, K2/K3
//   C/D: VGPR r -> lane<16: (M=r,  N=lane) ; lane>=16: (M=r+8, N=lane-16)
// So each lane's A and B fragments are contiguous float2's in row-major
// x[N,128] / W[128,128] (W row d holds the K axis -> B is W transposed, which
// is exactly x @ W.T).
// ---------------------------------------------------------------------------
__global__ void gcn_gemm_dual(const float* __restrict__ x,
                              const float* __restrict__ W,
                              const float* __restrict__ b,
                              const float* __restrict__ Ws,
                              const float* __restrict__ bs,
                              float* __restrict__ y,
                              float* __restrict__ out,
                              int n_nodes) {
  const int m0 = blockIdx.x * 16;
  if (m0 >= n_nodes) return;

  if (m0 + 16 > n_nodes) {
    // Block-uniform scalar fallback for a ragged tail (keeps WMMA branch EXEC==all 1s).
    const int rows = n_nodes - m0;
    for (int idx = threadIdx.x; idx < rows * DFEAT; idx += blockDim.x) {
      const int r = m0 + idx / DFEAT;
      const int d = idx % DFEAT;
      float accy = b[d], accs = bs[d];
      const float* xr = x + (size_t)r * DFEAT;
      const float* wr = W + (size_t)d * DFEAT;
      const float* wsr = Ws + (size_t)d * DFEAT;
      for (int k = 0; k < DFEAT; ++k) {
        accy = fmaf(xr[k], wr[k], accy);
        accs = fmaf(xr[k], wsr[k], accs);
      }
      y[(size_t)r * DFEAT + d] = accy;
      out[(size_t)r * DFEAT + d] = accs;
    }
    return;
  }

  const int lane = threadIdx.x & 31;
  const int wave = threadIdx.x >> 5;   // 0..7 -> column tile
  const int col0 = wave * 16;
  const int half = lane >> 4;          // 0: K pair {0,1}, 1: K pair {2,3}
  const int l16 = lane & 15;

  const float* xrow  = x  + (size_t)(m0 + l16) * DFEAT + half * 2;   // A fragment base
  const float* wrow  = W  + (size_t)(col0 + l16) * DFEAT + half * 2; // B (for y)
  const float* wsrow = Ws + (size_t)(col0 + l16) * DFEAT + half * 2; // B (for x2)

  v8f acc_y = {};
  v8f acc_s = {};

#pragma unroll 8
  for (int k = 0; k < DFEAT; k += 4) {
    v2f a  = *(const v2f*)(xrow + k);
    v2f by = *(const v2f*)(wrow + k);
    v2f bs2 = *(const v2f*)(wsrow + k);
    acc_y = __builtin_amdgcn_wmma_f32_16x16x4_f32(
        /*neg_a=*/false, a, /*neg_b=*/false, by,
        /*c_mod=*/(short)0, acc_y, /*reuse_a=*/false, /*reuse_b=*/false);
    acc_s = __builtin_amdgcn_wmma_f32_16x16x4_f32(
        /*neg_a=*/false, a, /*neg_b=*/false, bs2,
        /*c_mod=*/(short)0, acc_s, /*reuse_a=*/false, /*reuse_b=*/false);
  }

  const float bias_y = b[col0 + l16];
  const float bias_s = bs[col0 + l16];
#pragma unroll
  for (int r = 0; r < 8; ++r) {
    const int m = m0 + r + half * 8;
    const size_t idx = (size_t)m * DFEAT + col0 + l16;
    y[idx]   = acc_y[r] + bias_y;
    out[idx] = acc_s[r] + bias_s;
  }
}

// ---------------------------------------------------------------------------
// Kernel 2: edge gather-scale-scatter. One wave per edge; lane i handles the
// contiguous float4 at feature offset 4*i (32 lanes * 16B = 512B row).
// y and out are each 51.2MB -> resident in the 192MB L2, atomics execute at L2.
// ---------------------------------------------------------------------------
__global__ void gcn_edge_scatter(const int* __restrict__ erow,
                                 const int* __restrict__ ecol,
                                 const float* __restrict__ eval_,
                                 const float* __restrict__ y,
                                 float* __restrict__ out,
                                 int n_edges) {
  const int e = (int)((blockIdx.x * (unsigned)blockDim.x + threadIdx.x) >> 5);
  if (e >= n_edges) return;
  const int lane = threadIdx.x & 31;

  const int r = erow[e];
  const int c = ecol[e];
  const float v = eval_[e];

  const float4 m = *((const float4*)(y + (size_t)c * DFEAT) + lane);
  float* dst = out + (size_t)r * DFEAT + lane * 4;
  atomicAdd(dst + 0, v * m.x);
  atomicAdd(dst + 1, v * m.y);
  atomicAdd(dst + 2, v * m.z);
  atomicAdd(dst + 3, v * m.w);
}

// ---------------------------------------------------------------------------
// Kernel 3: in-place relu over out (n divisible by 4: D=128).
// ---------------------------------------------------------------------------
__global__ void gcn_relu(float* __restrict__ out, int n4) {
  const int i = blockIdx.x * blockDim.x + threadIdx.x;
  if (i >= n4) return;
  float4 v = ((float4*)out)[i];
  v.x = fmaxf(v.x, 0.0f);
  v.y = fmaxf(v.y, 0.0f);
  v.z = fmaxf(v.z, 0.0f);
  v.w = fmaxf(v.w, 0.0f);
  ((float4*)out)[i] = v;
}

extern "C" void kernel_launch(void* const* d_in, const int* in_sizes, int n_in,
                              void* d_out, int out_size, void* d_ws, size_t ws_size,
                              hipStream_t stream) {
  const float* x     = (const float*)d_in[0];
  const int*   erow  = (const int*)d_in[1];
  const int*   ecol  = (const int*)d_in[2];
  const float* eval_ = (const float*)d_in[3];
  const float* W     = (const float*)d_in[4];
  const float* b     = (const float*)d_in[5];
  const float* Ws    = (const float*)d_in[6];
  const float* bs    = (const float*)d_in[7];

  float* out = (float*)d_out;
  float* y   = (float*)d_ws;  // N*D floats = 51.2 MB scratch

  const int n_nodes = in_sizes[0] / DFEAT;
  const int n_edges = in_sizes[1];

  // 1) y = x@W.T + b ; out = x@Wself.T + bself
  const int row_tiles = (n_nodes + 15) / 16;
  gcn_gemm_dual<<<row_tiles, 256, 0, stream>>>(x, W, b, Ws, bs, y, out, n_nodes);

  // 2) out[row] += val * y[col]  (wave per edge, 8 edges per 256-thread block)
  const int edge_blocks = (n_edges + 7) / 8;
  gcn_edge_scatter<<<edge_blocks, 256, 0, stream>>>(erow, ecol, eval_, y, out, n_edges);

  // 3) out = relu(out)
  const int n4 = (n_nodes * DFEAT) / 4;
  gcn_relu<<<(n4 + 255) / 256, 256, 0, stream>>>(out, n4);
}